// GrapheneOptimizer_30623116821247
// MI455X (gfx1250) — compile-verified
//
#include <hip/hip_runtime.h>

typedef __attribute__((ext_vector_type(2))) float v2f;
typedef __attribute__((ext_vector_type(8))) float v8f;

#define NSEG_J 16
#define WAVES_PER_BLOCK 8
#define TILE_I (WAVES_PER_BLOCK * 16)   // 128 i-atoms per block

// ---------------------------------------------------------------------------
// Pass 1: all-pairs energy with WMMA-screened 16x16 tiles.
// Each wave owns a 16-row i-tile and sweeps a j-segment in 16-column chunks.
// v_wmma_f32_16x16x4_f32 computes the 16x16 Gram matrix G = Pi . Pj^T in one
// instruction; sq_est = |pi|^2 + |pj|^2 - 2G screens the chunk (margin 16.25
// vs true cutoff^2 = 16 absorbs f32 rounding). Surviving chunks run the exact
// per-pair transcendental math with the exact d < 4 mask, so screening cannot
// change the result.
// ---------------------------------------------------------------------------
__global__ __launch_bounds__(256) void graphene_energy_kernel(
    const float* __restrict__ pos,   // (N,2) row-major
    const float* __restrict__ ang,   // (N,)
    float* __restrict__ partial,     // one partial per block
    int n)
{
    __shared__ float red[256];

    const int tid  = threadIdx.x;
    const int wave = tid >> 5;
    const int lane = tid & 31;
    const int half = lane >> 4;   // 0: lanes 0-15, 1: lanes 16-31
    const int ln   = lane & 15;

    const int i0 = blockIdx.x * TILE_I + wave * 16;

    // A operand (16x4 f32): lanes 0-15 carry K=0,1 = (x,y); lanes 16-31 carry
    // K=2,3 which we zero (K-dim padded from 2 to 4).
    v2f A;
    {
        float ax = pos[2 * (i0 + ln) + 0];
        float ay = pos[2 * (i0 + ln) + 1];
        A.x = half ? 0.0f : ax;
        A.y = half ? 0.0f : ay;
    }

    // Rows of D this lane owns: M = r + 8*half (matches 16x16 f32 C/D layout).
    float pix[8], piy[8], ni[8];
#pragma unroll
    for (int r = 0; r < 8; ++r) {
        int m  = r + 8 * half;
        float x = pos[2 * (i0 + m) + 0];
        float y = pos[2 * (i0 + m) + 1];
        pix[r] = x;
        piy[r] = y;
        ni[r]  = x * x + y * y;
    }

    const int jseg = n / NSEG_J;            // 512, multiple of 16
    const int jbeg = blockIdx.y * jseg;
    const int jend = jbeg + jseg;

    const float PI_F = 3.14159265358979323846f;
    float acc = 0.0f;

    for (int j0 = jbeg; j0 < jend; j0 += 16) {
        float pjx = pos[2 * (j0 + ln) + 0];
        float pjy = pos[2 * (j0 + ln) + 1];
        float nj  = pjx * pjx + pjy * pjy;

        // B operand (4x16 f32): lanes 0-15 carry K=0,1 rows = (x,y); lanes
        // 16-31 carry K=2,3 rows which we zero.
        v2f B;
        B.x = half ? 0.0f : pjx;
        B.y = half ? 0.0f : pjy;

        v8f C = {0.f, 0.f, 0.f, 0.f, 0.f, 0.f, 0.f, 0.f};
        v8f G = __builtin_amdgcn_wmma_f32_16x16x4_f32(
            /*neg_a=*/false, A, /*neg_b=*/false, B,
            /*c_mod=*/(short)0, C, /*reuse_a=*/false, /*reuse_b=*/false);

        float mn = 1e30f;
#pragma unroll
        for (int r = 0; r < 8; ++r) {
            float s = (ni[r] + nj) - 2.0f * G[r];
            mn = fminf(mn, s);
        }

        // Wave-uniform cull: ~93% of chunks have no pair near the cutoff.
        if (__builtin_amdgcn_ballot_w32(mn < 16.25f) == 0u) continue;

        float angj = ang[j0 + ln];
#pragma unroll
        for (int r = 0; r < 8; ++r) {
            float vx = pjx - pix[r];
            float vy = pjy - piy[r];
            float sq = vx * vx + vy * vy;
            if (sq < 16.0f) {
                float d = (sq > 0.0f) ? sqrtf(sq) : 0.0f;
                if (d < 4.0f) {  // exact reference mask
                    float pa = atan2f(vy, vx + 1e-7f);
                    float aa = pa * 3.0f - angj;
                    float t  = d - 1.35f;
                    float eb = expf(-(t * t) / 0.3f);
                    float ca = cosf(aa);
                    float cb = cosf(aa + PI_F);
                    float e  = -eb * (expf(ca) - expf(cb))
                               + 7.0f * expf(-0.5f * d);
                    acc += e;
                }
            }
        }
    }

    // Deterministic block reduction.
    red[tid] = acc;
    __syncthreads();
    for (int s = 128; s > 0; s >>= 1) {
        if (tid < s) red[tid] += red[tid + s];
        __syncthreads();
    }
    if (tid == 0) partial[blockIdx.y * gridDim.x + blockIdx.x] = red[0];
}

// ---------------------------------------------------------------------------
// Pass 2: deterministic reduction of block partials into the scalar output.
// ---------------------------------------------------------------------------
__global__ __launch_bounds__(256) void graphene_reduce_kernel(
    const float* __restrict__ partial, int np, float* __restrict__ out)
{
    __shared__ float red[256];
    float s = 0.0f;
    for (int k = threadIdx.x; k < np; k += 256) s += partial[k];
    red[threadIdx.x] = s;
    __syncthreads();
    for (int t = 128; t > 0; t >>= 1) {
        if (threadIdx.x < t) red[threadIdx.x] += red[threadIdx.x + t];
        __syncthreads();
    }
    if (threadIdx.x == 0) out[0] = red[0];
}

extern "C" void kernel_launch(void* const* d_in, const int* in_sizes, int n_in,
                              void* d_out, int out_size, void* d_ws, size_t ws_size,
                              hipStream_t stream) {
    const float* pos = (const float*)d_in[0];   // positions (N,2) f32
    const float* ang = (const float*)d_in[1];   // angles (N,)  f32
    const int n = in_sizes[0] / 2;              // 8192

    float* partial = (float*)d_ws;              // nbx*NSEG_J floats (4 KB)
    const int nbx = n / TILE_I;                 // 64
    dim3 grid(nbx, NSEG_J);                     // 1024 blocks, 8 waves each

    graphene_energy_kernel<<<grid, 256, 0, stream>>>(pos, ang, partial, n);

    const int np = nbx * NSEG_J;
    graphene_reduce_kernel<<<1, 256, 0, stream>>>(partial, np, (float*)d_out);
}